// MyLinearRegression_66477503807564
// MI455X (gfx1250) — compile-verified
//
#include <hip/hip_runtime.h>

// y[i] = x[i,:] . w[0:32] + w[32]  ;  x: [N,32] f32, w: [33,1] f32, y: [N,1] f32
//
// HBM-bound: 512 MB of x streamed once (+16 MB y) -> ~23 us floor at 23.3 TB/s.
// Each wave32 owns exactly TILES consecutive 16-row tiles (launcher guarantees
// full waves; remainder rows go to a scalar tail kernel). Per tile: 8 x
// V_WMMA_F32_16X16X4_F32 in full f32, with B = w replicated across all 16
// columns so every accumulator column equals the answer. Wave id is forced
// scalar so all addressing is SGPR base + lane offset + immediate; the tile
// loop is fully unrolled so every VMEM offset is an instruction immediate.

typedef __attribute__((ext_vector_type(2))) float v2f;
typedef __attribute__((ext_vector_type(4))) float v4f;
typedef __attribute__((ext_vector_type(8))) float v8f;

#define DIM 32
#define TILES 8  // 16-row tiles per wave -> 128 rows / 16 KB of x per wave

__global__ __launch_bounds__(256) void linreg_wmma_f32(
    const float* __restrict__ x, const float* __restrict__ w,
    float* __restrict__ y) {
  const int lane = threadIdx.x & 31;
  // Force the wave id into an SGPR: uniform by construction.
  const int wib = __builtin_amdgcn_readfirstlane(threadIdx.x >> 5);
  const long long waveId = (long long)blockIdx.x * 8 + wib;
  const long long rowBase = waveId * (16LL * TILES);

  const int half = lane >> 4;  // 0: lanes 0-15 (K pair 0,1), 1: lanes 16-31 (2,3)
  const int m    = lane & 15;  // A-matrix row index held by this lane

  // B fragments, loaded once: B[k][n] = w[k] for all n.
  // VGPR0 holds K = 4j + 2*half, VGPR1 holds K = 4j + 2*half + 1.
  v2f bw[8];
#pragma unroll
  for (int j = 0; j < 8; ++j) {
    bw[j] = *(const v2f*)(w + 4 * j + 2 * half);  // tiny, cache-resident
  }
  const float bias = w[DIM];

  // Lane (m, half) streams row r at byte offsets 16*j + 8*half: b64 loads,
  // each 128 B row fully consumed by its lane pair -> 100% line utilization.
  const float* xrow = x + (size_t)rowBase * DIM + m * DIM + 2 * half;
  float*       yout = y + rowBase + 8 * half;

#pragma unroll
  for (int t = 0; t < TILES; ++t) {
    v8f c = {};
#pragma unroll
    for (int j = 0; j < 8; ++j) {
      // tile t at immediate byte offset 2048*t + 16*j (fits signed-24 imm)
      v2f a = __builtin_nontemporal_load(
          (const v2f*)(xrow + t * 16 * DIM + 4 * j));
      // (neg_a, A, neg_b, B, c_mod, C, reuse_a, reuse_b)
      c = __builtin_amdgcn_wmma_f32_16x16x4_f32(
          false, a, false, bw[j], (short)0, c, false, false);
    }

    // C/D layout: VGPR v, lanes 0-15 -> D[v][lane]; lanes 16-31 -> D[v+8][..].
    // Columns identical => lane 0 holds y[base+16t+0..7], lane 16 the rest.
    if (m == 0) {
      v4f lo = {c[0] + bias, c[1] + bias, c[2] + bias, c[3] + bias};
      v4f hi = {c[4] + bias, c[5] + bias, c[6] + bias, c[7] + bias};
      __builtin_nontemporal_store(lo, (v4f*)(yout + t * 16));
      __builtin_nontemporal_store(hi, (v4f*)(yout + t * 16 + 4));
    }
  }
}

// Generic scalar tail for rows not covered by full 128-row waves (0 blocks
// for N = 4,194,304; kept for correctness on arbitrary N).
__global__ void linreg_tail(const float* __restrict__ x,
                            const float* __restrict__ w,
                            float* __restrict__ y, int start, int nrows) {
  const int i = start + blockIdx.x * blockDim.x + threadIdx.x;
  if (i >= nrows) return;
  float acc = w[DIM];
#pragma unroll
  for (int k = 0; k < DIM; ++k) acc += x[(size_t)i * DIM + k] * w[k];
  y[i] = acc;
}

extern "C" void kernel_launch(void* const* d_in, const int* in_sizes, int n_in,
                              void* d_out, int out_size, void* d_ws, size_t ws_size,
                              hipStream_t stream) {
  const float* x = (const float*)d_in[0];
  const float* w = (const float*)d_in[1];
  float* y = (float*)d_out;

  const int nrows = in_sizes[0] / DIM;       // N = 4,194,304
  const int rowsPerBlock = 8 * 16 * TILES;   // 8 waves * 128 rows = 1024
  const int blocks = nrows / rowsPerBlock;   // 4096 full blocks

  if (blocks > 0)
    linreg_wmma_f32<<<blocks, 256, 0, stream>>>(x, w, y);

  const int covered = blocks * rowsPerBlock;
  const int rem = nrows - covered;           // 0 for this shape
  if (rem > 0) {
    const int tblocks = (rem + 255) / 256;
    linreg_tail<<<tblocks, 256, 0, stream>>>(x, w, y, covered, nrows);
  }
}